// CELWithLevenshteinRegularization_36936718745640
// MI455X (gfx1250) — compile-verified
//
#include <hip/hip_runtime.h>

#define B_ 4
#define S_ 128
#define V_ 2048
#define L_ 12
#define BS_ (B_ * S_)
#define REG_LAMBDA_ 0.01f
#define PK1 0x00010001u   // packed {1,1} in u16 halves

typedef __attribute__((ext_vector_type(2))) float v2f;
typedef __attribute__((ext_vector_type(8))) float v8f;

// V_PK_MIN_U16: packed unsigned-16 min (VOP3P). Not exposed by HIP headers.
__device__ __forceinline__ unsigned pk_min_u16(unsigned a, unsigned b) {
    unsigned d;
    asm("v_pk_min_u16 %0, %1, %2" : "=v"(d) : "v"(a), "v"(b));
    return d;
}

// Exact f32 wave-sum: D = ones(16x4) @ B(4x16) + 0 gives column sums of the 64
// B slots (32 lane partials + 32 zeros); summing D-VGPR0 over a 16-lane group
// yields the wave total regardless of B's lane/VGPR bijection.
__device__ __forceinline__ float wave_sum_wmma(float acc) {
    v2f a; a.x = 1.0f; a.y = 1.0f;
    v2f b; b.x = acc;  b.y = 0.0f;
    v8f c = {};
    c = __builtin_amdgcn_wmma_f32_16x16x4_f32(false, a, false, b, (short)0, c,
                                              false, false);
    float t = c[0];
#pragma unroll
    for (int off = 8; off >= 1; off >>= 1) t += __shfl_xor(t, off, 16);
    return t;  // valid in lane 0 of each 16-lane group
}

__global__ void zero_out_kernel(float* out) { out[0] = 0.0f; }

// --- Cross-entropy: one block per (b,s) row; exact log-softmax in f32. ---
__global__ __launch_bounds__(256) void ce_kernel(const float* __restrict__ pred,
                                                 const int* __restrict__ gold,
                                                 float* __restrict__ out) {
    __shared__ float smax[8];
    __shared__ float ssum[8];
    const int bs = blockIdx.x;
    const int tid = threadIdx.x;
    const int lane = tid & 31;
    const int wid = tid >> 5;
    const float* row = pred + (size_t)bs * V_;

    float lmax = -3.402823466e38f;
    for (int v = tid; v < V_; v += 256) lmax = fmaxf(lmax, row[v]);
#pragma unroll
    for (int off = 16; off >= 1; off >>= 1)
        lmax = fmaxf(lmax, __shfl_xor(lmax, off, 32));
    if (lane == 0) smax[wid] = lmax;
    __syncthreads();
    float gmax = smax[0];
#pragma unroll
    for (int i = 1; i < 8; ++i) gmax = fmaxf(gmax, smax[i]);

    float lsum = 0.0f;
    for (int v = tid; v < V_; v += 256) lsum += expf(row[v] - gmax);
    const float wsum = wave_sum_wmma(lsum);   // uniform flow: EXEC all ones
    if (lane == 0) ssum[wid] = wsum;
    __syncthreads();
    if (tid == 0) {
        float gsum = 0.0f;
#pragma unroll
        for (int i = 0; i < 8; ++i) gsum += ssum[i];
        const float nll = -(row[gold[bs]] - gmax - logf(gsum));
        atomicAdd(out, nll * (1.0f / (float)BS_));
    }
}

// --- Levenshtein + regularizer: one block per (b,s). Each thread runs 4
//     packed-u16 DPs, each DP solving TWO words (v and v+1024) at once.
//     DP values <= 25, so plain 32-bit adds never carry across halves;
//     3-way mins use V_PK_MIN_U16. ---
__global__ __launch_bounds__(256) void lev_reg_kernel(const float* __restrict__ pred,
                                                      const int* __restrict__ X,
                                                      const int* __restrict__ words,
                                                      float* __restrict__ out) {
    const int bs = blockIdx.x;
    const int tid = threadIdx.x;
    const float* prow = pred + (size_t)bs * V_;

    // Broadcast this sequence's 12 chars, duplicated into both u16 halves.
    unsigned x2[L_];
    const int* xrow = X + bs * L_;
#pragma unroll
    for (int i = 0; i < L_; ++i) {
        const unsigned xc = (unsigned)xrow[i];
        x2[i] = xc * PK1;
    }

    float acc = 0.0f;
    for (int v = tid; v < (V_ / 2); v += 256) {
        __builtin_prefetch(&prow[v], 0, 0);              // global_prefetch_b8
        __builtin_prefetch(&prow[v + V_ / 2], 0, 0);
        const int* w0 = words + v * L_;
        const int* w1 = words + (v + V_ / 2) * L_;
        unsigned wc2[L_];
#pragma unroll
        for (int i = 0; i < L_; ++i)
            wc2[i] = (unsigned)w0[i] | ((unsigned)w1[i] << 16);

        unsigned dp[L_ + 1];
#pragma unroll
        for (int j = 0; j <= L_; ++j) dp[j] = (unsigned)j * PK1;
#pragma unroll
        for (int k = 0; k < L_; ++k) {
            const unsigned c2 = wc2[k];
            unsigned diag = dp[0];
            dp[0] = (unsigned)(k + 1) * PK1;
            unsigned leftp1 = dp[0] + PK1;
#pragma unroll
            for (int j = 1; j <= L_; ++j) {
                const unsigned up = dp[j];
                const unsigned cost = pk_min_u16(x2[j - 1] ^ c2, PK1);
                const unsigned val =
                    pk_min_u16(pk_min_u16(up + PK1, leftp1), diag + cost);
                diag = up;
                dp[j] = val;
                leftp1 = val + PK1;
            }
        }
        const unsigned d2 = dp[L_];
        const int dlo = (int)(d2 & 0xFFFFu);
        const int dhi = (int)(d2 >> 16);
        const int wlo = dlo - 2;
        const int whi = dhi - 2;
        acc += (float)(wlo > 0 ? wlo : 0) * prow[v];
        acc += (float)(whi > 0 ? whi : 0) * prow[v + V_ / 2];
    }

    // Wave-level exact f32 reduction via WMMA, then LDS combine of 8 waves.
    const float wsum = wave_sum_wmma(acc);
    __shared__ float warpsum[8];
    const int lane = tid & 31;
    const int wid = tid >> 5;
    if (lane == 0) warpsum[wid] = wsum;
    __syncthreads();
    if (tid == 0) {
        float bsum = 0.0f;
#pragma unroll
        for (int i = 0; i < 8; ++i) bsum += warpsum[i];
        atomicAdd(out, bsum * REG_LAMBDA_);
    }
}

extern "C" void kernel_launch(void* const* d_in, const int* in_sizes, int n_in,
                              void* d_out, int out_size, void* d_ws, size_t ws_size,
                              hipStream_t stream) {
    const float* pred  = (const float*)d_in[0];
    const int*   gold  = (const int*)d_in[1];
    const int*   X     = (const int*)d_in[2];
    const int*   words = (const int*)d_in[3];
    float* out = (float*)d_out;

    zero_out_kernel<<<1, 1, 0, stream>>>(out);
    ce_kernel<<<BS_, 256, 0, stream>>>(pred, gold, out);
    lev_reg_kernel<<<BS_, 256, 0, stream>>>(pred, X, words, out);
}